// HybridBranch_87454124081675
// MI455X (gfx1250) — compile-verified
//
#include <hip/hip_runtime.h>
#include <hip/hip_bf16.h>

// Problem constants (from reference)
#define Hn  512   // hidden size
#define Cn  4096  // num classes
#define Tn  32    // time steps
#define Bn  64    // batch
#define Ln  256   // FH*FW spatial positions

typedef __attribute__((ext_vector_type(2))) float v2f;
typedef __attribute__((ext_vector_type(8))) float v8f;

// ---------------------------------------------------------------------------
// Kernel 1: ctx[b,h] = sum over 256 spatial positions of origin_fmap[b,h,:]
// One wave32 per (b,h) row: 2 coalesced float4 loads per lane, shuffle reduce.
// ---------------------------------------------------------------------------
__global__ void reduce_fmap_kernel(const float* __restrict__ fmap,
                                   float* __restrict__ ctx) {
    const int wave = (int)((blockIdx.x * blockDim.x + threadIdx.x) >> 5);
    const int lane = (int)(threadIdx.x & 31);
    if (wave >= Bn * Hn) return;
    const float4* p = (const float4*)(fmap + (size_t)wave * Ln);
    float4 a = p[lane];
    float4 b = p[lane + 32];
    float s = (a.x + a.y) + (a.z + a.w) + (b.x + b.y) + (b.z + b.w);
    #pragma unroll
    for (int m = 16; m >= 1; m >>= 1) s += __shfl_xor(s, m, 32);
    if (lane == 0) ctx[wave] = s;
}

// ---------------------------------------------------------------------------
// Kernel 2: D[64,4096] = ctx[64,512] @ W_gen^T + b_gen, via V_WMMA_F32_16X16X4_F32
// Block = 256 threads (8 waves). Block covers 128 N columns; each wave owns a
// 16-wide N tile across all 64 M rows (4 accumulator tiles). K staged through
// LDS in chunks of 32 (stride padded to 33 against bank conflicts).
// ---------------------------------------------------------------------------
#define KC  32
#define NB  128
#define LDP 33

__global__ void wmma_gemm_kernel(const float* __restrict__ ctx,
                                 const float* __restrict__ Wg,
                                 const float* __restrict__ bg,
                                 float* __restrict__ D) {
    __shared__ float As[Bn * LDP];   // 64 x 32 (+pad)
    __shared__ float Bs[NB * LDP];   // 128 x 32 (+pad)

    const int tid  = (int)threadIdx.x;
    const int wv   = tid >> 5;
    const int lane = tid & 31;
    const int l15  = lane & 15;
    const int half = lane >> 4;          // 0: lanes 0-15, 1: lanes 16-31
    const int n0   = (int)blockIdx.x * NB;
    const int nt   = wv * 16;            // this wave's N tile inside the block

    v8f acc[4] = {};                     // 4 M-tiles of 16x16 f32

    for (int k0 = 0; k0 < Hn; k0 += KC) {
        __syncthreads();
        // Stage A chunk: ctx[0:64, k0:k0+32] (coalesced: wave spans one row)
        #pragma unroll
        for (int i = 0; i < (Bn * KC) / 256; ++i) {
            int idx = tid + i * 256;
            int r = idx >> 5, c = idx & 31;
            As[r * LDP + c] = ctx[r * Hn + k0 + c];
        }
        // Stage B chunk: W_gen[n0:n0+128, k0:k0+32] (coalesced)
        #pragma unroll
        for (int i = 0; i < (NB * KC) / 256; ++i) {
            int idx = tid + i * 256;
            int r = idx >> 5, c = idx & 31;
            Bs[r * LDP + c] = Wg[(size_t)(n0 + r) * Hn + k0 + c];
        }
        __syncthreads();

        #pragma unroll
        for (int kk = 0; kk < KC; kk += 4) {
            // B 4x16 fragment: VGPR0 = K {kk, kk+2} per lane half, VGPR1 = +1
            v2f bf;
            bf.x = Bs[(nt + l15) * LDP + kk + 2 * half];
            bf.y = Bs[(nt + l15) * LDP + kk + 2 * half + 1];
            #pragma unroll
            for (int m = 0; m < 4; ++m) {
                // A 16x4 fragment: lane 0-15 -> K {kk,kk+1}; lane 16-31 -> {kk+2,kk+3}
                v2f af;
                af.x = As[(m * 16 + l15) * LDP + kk + 2 * half];
                af.y = As[(m * 16 + l15) * LDP + kk + 2 * half + 1];
                acc[m] = __builtin_amdgcn_wmma_f32_16x16x4_f32(
                    /*neg_a=*/false, af, /*neg_b=*/false, bf,
                    /*c_mod=*/(short)0, acc[m],
                    /*reuse_a=*/false, /*reuse_b=*/false);
            }
        }
    }

    // Epilogue: C/D layout — VGPR r: lanes 0-15 -> M=r, lanes 16-31 -> M=r+8
    const int n = n0 + nt + l15;
    const float bias = bg[n];
    #pragma unroll
    for (int m = 0; m < 4; ++m) {
        #pragma unroll
        for (int r = 0; r < 8; ++r) {
            int row = m * 16 + half * 8 + r;
            D[(size_t)row * Cn + n] = acc[m][r] + bias;
        }
    }
}

// ---------------------------------------------------------------------------
// Kernel 3: out[b,t,:] = D[b,:]  (broadcast over T). Fully coalesced float4.
// One block per (b,t); 256 threads x 4 float4 = 4096 floats.
// ---------------------------------------------------------------------------
__global__ void broadcast_kernel(const float* __restrict__ D,
                                 float* __restrict__ out) {
    const int bt = (int)blockIdx.x;     // 0..B*T-1
    const int b = bt >> 5;              // T == 32
    const int t = bt & 31;
    const float4* src = (const float4*)(D + (size_t)b * Cn);
    float4* dst = (float4*)(out + ((size_t)(b * Tn + t)) * Cn);
    #pragma unroll
    for (int i = 0; i < 4; ++i) {
        int idx = (int)threadIdx.x + i * 256;
        dst[idx] = src[idx];
    }
}

// ---------------------------------------------------------------------------
extern "C" void kernel_launch(void* const* d_in, const int* in_sizes, int n_in,
                              void* d_out, int out_size, void* d_ws, size_t ws_size,
                              hipStream_t stream) {
    // Input order: origin_fmap, text, is_train, W_emb, b_emb, W_ih, W_hh,
    //              b_ih, b_hh, W_gen, b_gen
    const float* fmap = (const float*)d_in[0];
    const float* Wg   = (const float*)d_in[9];
    const float* bg   = (const float*)d_in[10];
    float* out = (float*)d_out;

    float* ctx = (float*)d_ws;            // [64, 512]
    float* D   = ctx + (size_t)Bn * Hn;   // [64, 4096]

    // 1) spatial reduction: 32768 rows, 8 waves/block -> 4096 blocks
    reduce_fmap_kernel<<<(Bn * Hn) / 8, 256, 0, stream>>>(fmap, ctx);
    // 2) WMMA GEMM + bias: 4096/128 = 32 blocks
    wmma_gemm_kernel<<<Cn / NB, 256, 0, stream>>>(ctx, Wg, bg, D);
    // 3) broadcast over T: 64*32 = 2048 blocks
    broadcast_kernel<<<Bn * Tn, 256, 0, stream>>>(D, out);
}